// CustomRecall_78907139162810
// MI455X (gfx1250) — compile-verified
//
#include <hip/hip_runtime.h>
#include <hip/hip_bf16.h>

#define CLS          100
#define VEC_PER_ROW  25          // 100 floats = 25 x float4
#define TILE_ROWS    128
#define BLOCK        128
#define MAX_GRID     2048

// Issue async DMA copies of nVec float4s from global into LDS (exact layout
// copy). Does NOT wait: caller drains ASYNCcnt and barriers before reading.
__device__ __forceinline__ void async_issue(const float* __restrict__ gsrc,
                                            unsigned ldsBase, int nVec,
                                            int tid, int nthr) {
  unsigned long long gbase = (unsigned long long)(const void*)gsrc;
  for (int idx = tid; idx < nVec; idx += nthr) {
    unsigned           ldsoff = ldsBase + ((unsigned)idx << 4);
    unsigned long long gaddr  = gbase + ((unsigned long long)idx << 4);
    // Streaming read-once data: non-temporal so 800MB doesn't thrash L2/WGP$.
    asm volatile("global_load_async_to_lds_b128 %0, %1, off th:TH_LOAD_NT"
                 :: "v"(ldsoff), "v"(gaddr) : "memory");
  }
}

// First-occurrence argmax over one row of 100 floats held in LDS.
// Row stride = 25 float4; 25 mod 16 is odd => lanes hit distinct 4-bank
// groups per half-wave: conflict-free ds_load_b128.
__device__ __forceinline__ int row_argmax(const float* row) {
  const float4* rv = (const float4*)row;   // 16B-aligned (row stride 400B)
  float4 v = rv[0];
  float best = v.x; int bi = 0;
  if (v.y > best) { best = v.y; bi = 1; }
  if (v.z > best) { best = v.z; bi = 2; }
  if (v.w > best) { best = v.w; bi = 3; }
#pragma unroll
  for (int j = 1; j < VEC_PER_ROW; ++j) {
    v = rv[j];
    if (v.x > best) { best = v.x; bi = 4 * j;     }
    if (v.y > best) { best = v.y; bi = 4 * j + 1; }
    if (v.z > best) { best = v.z; bi = 4 * j + 2; }
    if (v.w > best) { best = v.w; bi = 4 * j + 3; }
  }
  return bi;
}

__global__ void __launch_bounds__(BLOCK)
recall_cm_kernel(const float* __restrict__ yt, const float* __restrict__ yp,
                 unsigned* __restrict__ g_tp, unsigned* __restrict__ g_ap,
                 int N) {
  __shared__ float    smem_t[TILE_ROWS * CLS];   // 51,200 B  y_true tile
  __shared__ float    smem_p[TILE_ROWS * CLS];   // 51,200 B  y_pred tile
  __shared__ unsigned s_tp[CLS];
  __shared__ unsigned s_ap[CLS];

  const int tid = threadIdx.x;
  if (tid < CLS) { s_tp[tid] = 0u; s_ap[tid] = 0u; }
  __syncthreads();

  // Low 32 bits of a generic pointer into the LDS aperture are the LDS byte
  // offset (ISA 10.2) — what async-to-LDS VDST wants.
  const unsigned ldsT = (unsigned)(unsigned long long)(void*)smem_t;
  const unsigned ldsP = (unsigned)(unsigned long long)(void*)smem_p;

  const int numTiles = (N + TILE_ROWS - 1) / TILE_ROWS;
  for (int tile = blockIdx.x; tile < numTiles; tile += gridDim.x) {
    const int    rowBase  = tile * TILE_ROWS;
    const int    rows     = min(TILE_ROWS, N - rowBase);
    const int    nVec     = rows * VEC_PER_ROW;
    const size_t elemBase = (size_t)rowBase * CLS;

    // Issue BOTH matrices' DMA streams before draining: ~50 b128 async ops
    // in flight per thread, one wait + one barrier per tile.
    async_issue(yt + elemBase, ldsT, nVec, tid, BLOCK);
    async_issue(yp + elemBase, ldsP, nVec, tid, BLOCK);
    asm volatile("s_wait_asynccnt 0x0" ::: "memory");
    __syncthreads();                     // all waves' copies visible

    if (tid < rows) {
      const int t = row_argmax(&smem_t[tid * CLS]);
      const int p = row_argmax(&smem_p[tid * CLS]);
      atomicAdd(&s_ap[t], 1u);
      if (p == t) atomicAdd(&s_tp[t], 1u);
    }
    __syncthreads();                     // reads done before next tile's DMA
  }

  // One flush per block of the per-class partial counters.
  if (tid < CLS) {
    unsigned a  = s_ap[tid];
    unsigned tv = s_tp[tid];
    if (a)  atomicAdd(&g_ap[tid], a);
    if (tv) atomicAdd(&g_tp[tid], tv);
  }
}

__global__ void zero_counters_kernel(unsigned* __restrict__ ws) {
  int i = blockIdx.x * blockDim.x + threadIdx.x;
  if (i < 2 * CLS) ws[i] = 0u;
}

__global__ void finalize_kernel(const unsigned* __restrict__ g_tp,
                                const unsigned* __restrict__ g_ap,
                                float* __restrict__ out) {
  __shared__ float red[128];
  const int tid = threadIdx.x;                     // 128 threads
  float r = 0.0f;
  if (tid < CLS) {
    // eps = float32 machine epsilon, matching jnp.finfo(float32).eps
    r = (float)g_tp[tid] / ((float)g_ap[tid] + 1.1920928955078125e-07f);
  }
  red[tid] = r;
  __syncthreads();
  for (int s = 64; s > 0; s >>= 1) {
    if (tid < s) red[tid] += red[tid + s];
    __syncthreads();
  }
  if (tid == 0) out[0] = red[0] * (1.0f / (float)CLS);
}

extern "C" void kernel_launch(void* const* d_in, const int* in_sizes, int n_in,
                              void* d_out, int out_size, void* d_ws, size_t ws_size,
                              hipStream_t stream) {
  const float* yt = (const float*)d_in[0];   // y_true [N, 100] fp32
  const float* yp = (const float*)d_in[1];   // y_pred [N, 100] fp32
  const int N = in_sizes[0] / CLS;           // 1,000,000

  unsigned* ws   = (unsigned*)d_ws;
  unsigned* g_tp = ws;                       // [100]
  unsigned* g_ap = ws + CLS;                 // [100]

  zero_counters_kernel<<<1, 256, 0, stream>>>(ws);

  const int numTiles = (N + TILE_ROWS - 1) / TILE_ROWS;
  const int grid     = numTiles < MAX_GRID ? numTiles : MAX_GRID;
  recall_cm_kernel<<<grid, BLOCK, 0, stream>>>(yt, yp, g_tp, g_ap, N);

  finalize_kernel<<<1, 128, 0, stream>>>(g_tp, g_ap, (float*)d_out);
}